// AdaptiveAutoCorrelation_85298050499040
// MI455X (gfx1250) — compile-verified
//
#include <hip/hip_runtime.h>
#include <math.h>

// ---------------------------------------------------------------------------
// AdaptiveAutoCorrelation on MI455X (gfx1250, wave32).
// FFTs expressed as real-DFT GEMMs executed on V_WMMA_F32_16X16X4_F32,
// with the shared B panel staged into LDS by the Tensor Data Mover
// (TENSOR_LOAD_TO_LDS + S_WAIT_TENSORCNT).
// Shapes fixed by the reference: B=8, L=1536, H=8, E=64 -> S = B*H*E = 4096.
// ---------------------------------------------------------------------------

typedef __attribute__((ext_vector_type(2))) float        v2f;
typedef __attribute__((ext_vector_type(8))) float        v8f;
typedef __attribute__((ext_vector_type(4))) unsigned int v4u;
typedef __attribute__((ext_vector_type(8))) int          v8i;
typedef __attribute__((ext_vector_type(4))) int          v4i;

#define BB 8
#define LL 1536
#define HH 8
#define EE 64
#define SS 4096            // B*H*E sequences
#define TOPK 7             // int(log(1536)) == 7
#define PI2 6.28318530717958647692f
#define KT 64              // K panel staged in LDS per block iteration

// ---------------------------------------------------------------------------
// TDM: DMA a [tile_h x 16] fp32 tile (row stride = stride_elems) from global
// memory into LDS at byte offset lds_byte_off. 2-D descriptor per ISA §8.3-8.5;
// groups 2/3 zero (<=2D tensor). Issued by one wave; tracked by TENSORcnt.
// ---------------------------------------------------------------------------
__device__ __forceinline__ void tdm_load_panel16(unsigned lds_byte_off,
                                                 const float* gptr,
                                                 unsigned tile_h,
                                                 unsigned stride_elems) {
  unsigned long long ga = (unsigned long long)(uintptr_t)gptr;
  v4u g0;
  g0[0] = 1u;                                     // count=1 (valid), user mode
  g0[1] = lds_byte_off;                           // [63:32] lds_addr
  g0[2] = (unsigned)(ga & 0xFFFFFFFFu);           // [95:64] global_addr lo
  g0[3] = ((unsigned)(ga >> 32) & 0x01FFFFFFu)    // [120:96] global_addr hi
          | (2u << 30);                           // [127:126] type = 2 (image)
  v8i g1;
  g1[0] = (int)(2u << 16);                        // data_size = 2 (4 bytes)
  g1[1] = (int)((16u & 0xFFFFu) << 16);           // tensor_dim0[15:0] = 16
  g1[2] = (int)((tile_h & 0xFFFFu) << 16);        // dim0 hi=0 | tensor_dim1 lo
  g1[3] = (int)((16u & 0xFFFFu) << 16);           // dim1 hi=0 | tile_dim0 = 16
  g1[4] = (int)(tile_h & 0xFFFFu);                // tile_dim1 | tile_dim2 = 0
  g1[5] = (int)stride_elems;                      // tensor_dim0_stride[31:0]
  g1[6] = 0;                                      // stride hi | dim1_stride lo
  g1[7] = 0;
  v4i z4 = {};
  v8i z8 = {};
  __builtin_amdgcn_tensor_load_to_lds(g0, g1, z4, z4, z8, 0);
}

// ---------------------------------------------------------------------------
// Stage A: LayerNorm over E (=64) and transpose (B,L,H,E) -> [seq=(b*H+h)*E+e][l]
// One wave per (b,l,h) row: lane holds e=lane and e=lane+32.
// ---------------------------------------------------------------------------
__global__ void __launch_bounds__(256) k_ln_t(const float* __restrict__ x,
                                              float* __restrict__ y) {
  int gwave = (int)((blockIdx.x * blockDim.x + threadIdx.x) >> 5);
  int lane  = threadIdx.x & 31;
  int total = BB * LL * HH;
  if (gwave >= total) return;
  int h = gwave % HH;
  int t = gwave / HH;
  int l = t % LL;
  int b = t / LL;
  const float* row = x + (size_t)gwave * EE;
  float x0 = row[lane];
  float x1 = row[lane + 32];
  float s  = x0 + x1;
  float s2 = x0 * x0 + x1 * x1;
#pragma unroll
  for (int off = 16; off > 0; off >>= 1) {
    s  += __shfl_xor(s,  off, 32);
    s2 += __shfl_xor(s2, off, 32);
  }
  float mean = s * (1.0f / EE);
  float var  = s2 * (1.0f / EE) - mean * mean;
  float inv  = rsqrtf(var + 1e-5f);
  size_t base = ((size_t)(b * HH + h) * EE) * LL + l;
  y[base + (size_t)lane * LL]        = (x0 - mean) * inv;
  y[base + (size_t)(lane + 32) * LL] = (x1 - mean) * inv;
}

// ---------------------------------------------------------------------------
// Downsample along time by mean-pooling `factor` samples.
// ---------------------------------------------------------------------------
__global__ void k_down(const float* __restrict__ src, float* __restrict__ dst,
                       int Nin, int Nout, int factor) {
  size_t i = (size_t)blockIdx.x * blockDim.x + threadIdx.x;
  size_t total = (size_t)SS * Nout;
  if (i >= total) return;
  int j = (int)(i % Nout);
  size_t seq = i / Nout;
  const float* r = src + seq * (size_t)Nin + (size_t)j * factor;
  float acc = 0.f;
  for (int t = 0; t < factor; ++t) acc += r[t];
  dst[i] = acc / (float)factor;
}

// ---------------------------------------------------------------------------
// Forward DFT basis: Fc[n,m]=cos(2*pi*n*m/N), Fs[n,m]=-sin(2*pi*n*m/N); m>=M -> 0
// ---------------------------------------------------------------------------
__global__ void k_basis_fwd(float* __restrict__ Fc, float* __restrict__ Fs,
                            int N, int M, int Mf) {
  size_t i = (size_t)blockIdx.x * blockDim.x + threadIdx.x;
  size_t total = (size_t)N * Mf;
  if (i >= total) return;
  int m = (int)(i % Mf);
  int n = (int)(i / Mf);
  if (m >= M) { Fc[i] = 0.f; Fs[i] = 0.f; return; }
  unsigned r = (unsigned)(((unsigned long long)n * (unsigned)m) % (unsigned)N);
  float ang = (float)r * (PI2 / (float)N);
  float sv, cv;
  __sincosf(ang, &sv, &cv);
  Fc[i] = cv;
  Fs[i] = -sv;
}

// ---------------------------------------------------------------------------
// Inverse rDFT basis with Hermitian weights folded in:
// Gc[m,n] = w_m*cos(2*pi*m*n/N)/N, Gs[m,n] = -w_m*sin(..)/N, w = 1 for m in
// {0, N/2} else 2; rows m>=M zeroed so K-padding contributes nothing.
// ---------------------------------------------------------------------------
__global__ void k_basis_inv(float* __restrict__ Gc, float* __restrict__ Gs,
                            int N, int M, int Mf) {
  size_t i = (size_t)blockIdx.x * blockDim.x + threadIdx.x;
  size_t total = (size_t)Mf * N;
  if (i >= total) return;
  int n = (int)(i % N);
  int m = (int)(i / N);
  if (m >= M) { Gc[i] = 0.f; Gs[i] = 0.f; return; }
  float w = (m == 0 || 2 * m == N) ? 1.f : 2.f;
  unsigned r = (unsigned)(((unsigned long long)n * (unsigned)m) % (unsigned)N);
  float ang = (float)r * (PI2 / (float)N);
  float sv, cv;
  __sincosf(ang, &sv, &cv);
  float sc = w / (float)N;
  Gc[i] = cv * sc;
  Gs[i] = -sv * sc;
}

// ---------------------------------------------------------------------------
// fp32 WMMA GEMM:  C[M x N] = A0[M x K0]*B0[K0 x N]  (+ A1[M x K1]*B1[K1 x N])
// Row-major, tight strides. Block = 8 waves; each wave owns one 16x16 C tile
// (8 row-tiles x same 16-col tile). The shared B K-panel is DMA'd into LDS by
// the Tensor Data Mover (wave 0 issues, S_WAIT_TENSORCNT, then barrier), and
// WMMA B fragments are fed from LDS (8x less global traffic). A stays in
// global (per-wave rows, L2-resident) with a prefetch one panel ahead.
// Requires M%128==0, N%16==0, K%4==0 (all satisfied; K%KT may be nonzero).
// Fragment layouts per CDNA5 ISA 7.12.2.
// ---------------------------------------------------------------------------
__global__ void __launch_bounds__(256)
k_gemm2_wmma(const float* __restrict__ A0, const float* __restrict__ B0, int K0,
             const float* __restrict__ A1, const float* __restrict__ B1, int K1,
             float* __restrict__ C, int M, int N) {
  __shared__ float ldsB[KT * 16];
  const int lane = threadIdx.x & 31;
  const int wave = threadIdx.x >> 5;
  const int row  = ((int)blockIdx.y * 8 + wave) * 16;
  const int col  = (int)blockIdx.x * 16;
  const int lh   = lane >> 4;   // 0: K pair {0,1}, 1: K pair {2,3}
  const int ll   = lane & 15;
  const unsigned ldsOff = (unsigned)(uintptr_t)(&ldsB[0]);  // LDS byte offset

  v8f acc = {};

  for (int pass = 0; pass < 2; ++pass) {
    const float* A = pass ? A1 : A0;   // kernel args: block-uniform
    const float* B = pass ? B1 : B0;
    const int    K = pass ? K1 : K0;
    if (A == nullptr) break;
    const float* Ar = A + (size_t)(row + ll) * K;

    for (int k0 = 0; k0 < K; k0 += KT) {
      const int chunk = (K - k0 < KT) ? (K - k0) : KT;
      __syncthreads();  // previous panel fully consumed
      if (wave == 0) {
        tdm_load_panel16(ldsOff, B + (size_t)k0 * N + col,
                         (unsigned)chunk, (unsigned)N);
        __builtin_amdgcn_s_wait_tensorcnt(0);
      }
      __syncthreads();  // panel visible to all waves
      if (k0 + KT < K)  // prefetch next A panel into caches
        __builtin_prefetch(Ar + k0 + KT, 0, 3);
#pragma unroll 4
      for (int k = 0; k < chunk; k += 4) {
        const int kk = k + lh * 2;
        v2f a = *(const v2f*)(Ar + k0 + kk);
        v2f b;
        b.x = ldsB[kk * 16 + ll];
        b.y = ldsB[(kk + 1) * 16 + ll];
        acc = __builtin_amdgcn_wmma_f32_16x16x4_f32(false, a, false, b,
                                                    (short)0, acc, false, false);
      }
    }
  }

  float* Cp = C + (size_t)(row + lh * 8) * N + col + ll;
#pragma unroll
  for (int j = 0; j < 8; ++j) Cp[(size_t)j * N] = acc[j];
}

// ---------------------------------------------------------------------------
// Normalized cross-spectrum, in place into (Qre,Qim):
//   qf = f*Q, kf = f*K;  res = qf*conj(kf)/(|kf|+eps)
// ---------------------------------------------------------------------------
__global__ void k_spec(float* __restrict__ Qre, float* __restrict__ Qim,
                       const float* __restrict__ Kre, const float* __restrict__ Kim,
                       const float* __restrict__ ff, int Mf, int M) {
  size_t i = (size_t)blockIdx.x * blockDim.x + threadIdx.x;
  size_t total = (size_t)SS * Mf;
  if (i >= total) return;
  int m = (int)(i % Mf);
  float f = 1.f / (1.f + __expf(-ff[0]));
  float qr = Qre[i], qi = Qim[i], kr = Kre[i], ki = Kim[i];
  float denom = f * sqrtf(kr * kr + ki * ki) + 1e-8f;
  float sc = f * f / denom;
  float rr = (qr * kr + qi * ki) * sc;
  float ri = (qi * kr - qr * ki) * sc;
  if (m >= M) { rr = 0.f; ri = 0.f; }
  Qre[i] = rr;
  Qim[i] = ri;
}

// ---------------------------------------------------------------------------
// corr (+)= softmax(scale_weights)[si] * lin_interp(c, L)
// ---------------------------------------------------------------------------
__global__ void k_accum(float* __restrict__ corr, const float* __restrict__ c,
                        int Ns, const float* __restrict__ sw, int si, int first) {
  size_t i = (size_t)blockIdx.x * blockDim.x + threadIdx.x;
  size_t total = (size_t)SS * LL;
  if (i >= total) return;
  int l = (int)(i % LL);
  size_t seq = i / LL;
  float w0 = sw[0], w1 = sw[1], w2 = sw[2];
  float mx = fmaxf(w0, fmaxf(w1, w2));
  float e0 = __expf(w0 - mx), e1 = __expf(w1 - mx), e2 = __expf(w2 - mx);
  float wsel = ((si == 0) ? e0 : (si == 1) ? e1 : e2) / (e0 + e1 + e2);
  float val;
  if (Ns == LL) {
    val = c[seq * (size_t)LL + l];
  } else {
    float scale = (float)Ns / (float)LL;
    float coord = ((float)l + 0.5f) * scale - 0.5f;
    coord = fminf(fmaxf(coord, 0.f), (float)(Ns - 1));
    int lo = (int)floorf(coord);
    int hi = lo + 1;
    if (hi > Ns - 1) hi = Ns - 1;
    float fr = coord - (float)lo;
    const float* rowp = c + seq * (size_t)Ns;
    val = rowp[lo] * (1.f - fr) + rowp[hi] * fr;
  }
  float add = wsel * val;
  corr[i] = first ? add : (corr[i] + add);
}

// ---------------------------------------------------------------------------
// mean over (h,e) of clip(corr, -10, 10):  meanc[b, l]
// ---------------------------------------------------------------------------
__global__ void __launch_bounds__(256) k_mean(const float* __restrict__ corr,
                                              float* __restrict__ meanc) {
  int l = (int)blockIdx.y * 256 + (int)threadIdx.x;
  int b = (int)blockIdx.x;
  float s = 0.f;
  for (int si = 0; si < HH * EE; ++si) {
    float v = corr[((size_t)(b * HH * EE + si)) * LL + l];
    v = fminf(fmaxf(v, -10.f), 10.f);
    s += v;
  }
  meanc[b * LL + l] = s * (1.f / (float)(HH * EE));
}

// ---------------------------------------------------------------------------
// Per-batch top-7 over L values + softmax over the 7 peaks.
// ---------------------------------------------------------------------------
__global__ void __launch_bounds__(256) k_topk(const float* __restrict__ meanc,
                                              float* __restrict__ nw,
                                              int* __restrict__ delays) {
  __shared__ float vals[LL];
  __shared__ float rmax[256];
  __shared__ int   ridx[256];
  __shared__ float wk[TOPK];
  __shared__ int   dk[TOPK];
  int b = (int)blockIdx.x;
  int t = (int)threadIdx.x;
  for (int i = t; i < LL; i += 256) vals[i] = meanc[b * LL + i];
  __syncthreads();
  for (int k = 0; k < TOPK; ++k) {
    float bv = -1e30f;
    int   bi = 0;
    for (int i = t; i < LL; i += 256) {
      if (vals[i] > bv) { bv = vals[i]; bi = i; }
    }
    rmax[t] = bv;
    ridx[t] = bi;
    __syncthreads();
    for (int off = 128; off > 0; off >>= 1) {
      if (t < off) {
        bool take = (rmax[t + off] > rmax[t]) ||
                    (rmax[t + off] == rmax[t] && ridx[t + off] < ridx[t]);
        if (take) { rmax[t] = rmax[t + off]; ridx[t] = ridx[t + off]; }
      }
      __syncthreads();
    }
    if (t == 0) {
      wk[k] = rmax[0];
      dk[k] = ridx[0];
      vals[ridx[0]] = -1e30f;
    }
    __syncthreads();
  }
  if (t == 0) {
    float mx = wk[0];
    for (int k = 1; k < TOPK; ++k) mx = fmaxf(mx, wk[k]);
    float se = 0.f, ex[TOPK];
    for (int k = 0; k < TOPK; ++k) { ex[k] = __expf(wk[k] - mx); se += ex[k]; }
    for (int k = 0; k < TOPK; ++k) {
      nw[b * TOPK + k]     = ex[k] / se;
      delays[b * TOPK + k] = dk[k];
    }
  }
}

// ---------------------------------------------------------------------------
// out[b,l,h,e] = sum_k nw[b,k] * values[b, (l+delay_k) mod L, h, e]
// block = one (b,l); 512 threads cover (h,e) contiguously (coalesced).
// ---------------------------------------------------------------------------
__global__ void __launch_bounds__(512) k_agg(const float* __restrict__ values,
                                             const float* __restrict__ nw,
                                             const int* __restrict__ delays,
                                             float* __restrict__ out) {
  int bl = (int)blockIdx.x;
  int l = bl % LL;
  int b = bl / LL;
  int t = (int)threadIdx.x;  // h*64 + e
  float acc = 0.f;
#pragma unroll
  for (int k = 0; k < TOPK; ++k) {
    int d = delays[b * TOPK + k];
    int lk = l + d;
    if (lk >= LL) lk -= LL;
    acc += nw[b * TOPK + k] * values[((size_t)b * LL + lk) * (HH * EE) + t];
  }
  out[((size_t)b * LL + l) * (HH * EE) + t] = acc;
}

// ---------------------------------------------------------------------------
// Host orchestration (all launches on `stream`; graph-capture safe).
// ---------------------------------------------------------------------------
extern "C" void kernel_launch(void* const* d_in, const int* in_sizes, int n_in,
                              void* d_out, int out_size, void* d_ws, size_t ws_size,
                              hipStream_t stream) {
  const float* q    = (const float*)d_in[0];
  const float* kkey = (const float*)d_in[1];
  const float* vals = (const float*)d_in[2];
  const float* sw   = (const float*)d_in[3];
  const float* ff   = (const float*)d_in[4];
  float* out = (float*)d_out;

  // ---- workspace carve-up (floats) ----
  const size_t SZ_SEQ_L = (size_t)SS * LL;       // 6,291,456
  const size_t SZ_SEQ_H = (size_t)SS * (LL / 2); // 3,145,728
  const size_t MF1 = 784;                        // pad16(1536/2+1)
  const size_t SZ_SPEC = (size_t)SS * MF1;       // 3,211,264
  const size_t SZ_BAS  = (size_t)LL * MF1;       // 1,204,224

  float* ws = (float*)d_ws;
  size_t off = 0;
  auto alloc = [&](size_t n) { float* p = ws + off; off += n; return p; };
  float* qn   = alloc(SZ_SEQ_L);
  float* kn   = alloc(SZ_SEQ_L);
  float* qd   = alloc(SZ_SEQ_H);
  float* kd   = alloc(SZ_SEQ_H);
  float* Qre  = alloc(SZ_SPEC);
  float* Qim  = alloc(SZ_SPEC);
  float* Kre  = alloc(SZ_SPEC);
  float* Kim  = alloc(SZ_SPEC);
  float* Fc   = alloc(SZ_BAS);
  float* Fs   = alloc(SZ_BAS);
  float* Gc   = alloc(SZ_BAS);
  float* Gs   = alloc(SZ_BAS);
  float* cbuf = alloc(SZ_SEQ_L);
  float* corr = alloc(SZ_SEQ_L);
  float* meanc = alloc((size_t)BB * LL);
  float* nwb   = alloc(64);
  int*   dly   = (int*)alloc(64);
  if (off * sizeof(float) > ws_size) return;  // not enough scratch: no-op

  const int TPB = 256;

  // Stage A: layernorm + transpose
  {
    int blocks = (BB * LL * HH + 7) / 8;  // 8 waves per block
    k_ln_t<<<blocks, TPB, 0, stream>>>(q, qn);
    k_ln_t<<<blocks, TPB, 0, stream>>>(kkey, kn);
  }

  // Per-scale spectral correlation via WMMA GEMMs
  for (int si = 0; si < 3; ++si) {
    const int s  = 1 << si;
    const int Ns = LL / s;                // 1536 / 768 / 384
    const int M  = Ns / 2 + 1;            // 769 / 385 / 193
    const int Mf = ((M + 15) / 16) * 16;  // 784 / 400 / 208
    const float* qs;
    const float* ks;
    if (si == 0) {
      qs = qn; ks = kn;
    } else {
      size_t tot = (size_t)SS * Ns;
      int g = (int)((tot + TPB - 1) / TPB);
      k_down<<<g, TPB, 0, stream>>>(qn, qd, LL, Ns, s);
      k_down<<<g, TPB, 0, stream>>>(kn, kd, LL, Ns, s);
      qs = qd; ks = kd;
    }

    {
      size_t tf = (size_t)Ns * Mf;
      k_basis_fwd<<<(int)((tf + TPB - 1) / TPB), TPB, 0, stream>>>(Fc, Fs, Ns, M, Mf);
      size_t ti = (size_t)Mf * Ns;
      k_basis_inv<<<(int)((ti + TPB - 1) / TPB), TPB, 0, stream>>>(Gc, Gs, Ns, M, Mf);
    }

    // Forward rDFT: [4096 x Ns] x [Ns x Mf]
    dim3 gF(Mf / 16, SS / 128);
    k_gemm2_wmma<<<gF, TPB, 0, stream>>>(qs, Fc, Ns, nullptr, nullptr, 0, Qre, SS, Mf);
    k_gemm2_wmma<<<gF, TPB, 0, stream>>>(qs, Fs, Ns, nullptr, nullptr, 0, Qim, SS, Mf);
    k_gemm2_wmma<<<gF, TPB, 0, stream>>>(ks, Fc, Ns, nullptr, nullptr, 0, Kre, SS, Mf);
    k_gemm2_wmma<<<gF, TPB, 0, stream>>>(ks, Fs, Ns, nullptr, nullptr, 0, Kim, SS, Mf);

    // Normalized cross-spectrum (in place into Qre/Qim)
    {
      size_t tot = (size_t)SS * Mf;
      k_spec<<<(int)((tot + TPB - 1) / TPB), TPB, 0, stream>>>(Qre, Qim, Kre, Kim, ff, Mf, M);
    }

    // Inverse rDFT as fused dual GEMM: cbuf = Xre*Gc + Xim*Gs
    dim3 gI(Ns / 16, SS / 128);
    k_gemm2_wmma<<<gI, TPB, 0, stream>>>(Qre, Gc, Mf, Qim, Gs, Mf, cbuf, SS, Ns);

    // Interp to L and accumulate with softmax(scale_weights)
    {
      size_t tot = (size_t)SS * LL;
      k_accum<<<(int)((tot + TPB - 1) / TPB), TPB, 0, stream>>>(corr, cbuf, Ns, sw, si, si == 0 ? 1 : 0);
    }
  }

  // clip + mean over (h,e)
  k_mean<<<dim3(BB, LL / 256), 256, 0, stream>>>(corr, meanc);
  // top-7 + softmax per batch
  k_topk<<<BB, 256, 0, stream>>>(meanc, nwb, dly);
  // delayed aggregation of values
  k_agg<<<BB * LL, 512, 0, stream>>>(vals, nwb, dly, out);
}